// OAMPNet_54674933678596
// MI455X (gfx1250) — compile-verified
//
#include <hip/hip_runtime.h>
#include <math.h>

typedef float v2f __attribute__((ext_vector_type(2)));
typedef float v8f __attribute__((ext_vector_type(8)));

#define EPSV 1e-9f

__device__ __forceinline__ v8f wmma_f32_4(v2f a, v2f b, v8f c) {
  // 8-arg pattern: (neg_a, A, neg_b, B, c_mod, C, reuse_a, reuse_b)
  return __builtin_amdgcn_wmma_f32_16x16x4_f32(false, a, false, b, (short)0, c,
                                               false, false);
}

__device__ __forceinline__ float waveReduce(float v) {
  #pragma unroll
  for (int o = 16; o > 0; o >>= 1) v += __shfl_xor(v, o, 32);
  return v;
}

__global__ __launch_bounds__(256)
void oampnet_kernel(const float* __restrict__ gH, const float* __restrict__ gy,
                    const float* __restrict__ gsig,
                    const float* __restrict__ ggam, const float* __restrict__ gth,
                    float* __restrict__ gx, int T) {
  const int b    = blockIdx.x;
  const int tid  = threadIdx.x;
  const int lane = tid & 31;
  const int wave = tid >> 5;

  __shared__ float sH[64 * 32];    // H        (row-major M x N)
  __shared__ float sHHt[64 * 64];  // H H^T    (reused every iteration)
  __shared__ float sA[64 * 64];    // A -> Cholesky L in-place (lower)
  __shared__ float sS[64 * 32];    // H -> A^{-1} H in-place
  __shared__ float sG[32 * 32];    // S^T H
  __shared__ float sy[64];
  __shared__ float sz[64];
  __shared__ float sx[32];
  __shared__ float sr[32];
  __shared__ float sDi[64];        // 1 / L[k][k]
  __shared__ float sred[4];        // reduction accumulators

  // ---------------- Phase 0: load H, y; trHtH and ||y||^2 ----------------
  const float* Hg = gH + (size_t)b * 2048;
  float trh = 0.f;
  for (int i = tid; i < 2048; i += 256) {
    float h = Hg[i];
    sH[i] = h;
    trh += h * h;
  }
  float yn = 0.f;
  if (tid < 64) {
    float v = gy[(size_t)b * 64 + tid];
    sy[tid] = v;
    yn = v * v;
  }
  if (tid < 32) sx[tid] = 0.f;
  if (tid == 0) { sred[0] = 0.f; sred[1] = 0.f; sred[2] = 0.f; sred[3] = 0.f; }
  __syncthreads();
  trh = waveReduce(trh);
  yn  = waveReduce(yn);
  if (lane == 0) {
    atomicAdd(&sred[0], trh);
    if (wave < 2) atomicAdd(&sred[1], yn);
  }
  __syncthreads();
  const float sigma2 = gsig[b];
  const float trHtH  = sred[0];
  float v2 = fmaxf((sred[1] - 64.f * sigma2) / trHtH, EPSV);

  // ---------------- Phase 1: HHt = H @ H^T via fp32 WMMA ------------------
  {
    const int m  = lane & 15;
    const int kh = (lane >> 4) * 2;   // K offset 0 or 2 inside the 4-chunk
    for (int t = wave; t < 16; t += 8) {   // 4x4 grid of 16x16 tiles
      const int ti = (t >> 2) * 16;
      const int tj = (t & 3) * 16;
      v8f acc = {};
      for (int k0 = 0; k0 < 32; k0 += 4) {
        v2f a, bb;
        a.x  = sH[(ti + m) * 32 + k0 + kh + 0];   // A[m][k]   = H[ti+m][k]
        a.y  = sH[(ti + m) * 32 + k0 + kh + 1];
        bb.x = sH[(tj + m) * 32 + k0 + kh + 0];   // B[k][n]   = H[tj+n][k]
        bb.y = sH[(tj + m) * 32 + k0 + kh + 1];
        acc = wmma_f32_4(a, bb, acc);
      }
      const int half = lane >> 4, n = lane & 15;
      #pragma unroll
      for (int v = 0; v < 8; ++v)
        sHHt[(ti + v + half * 8) * 64 + tj + n] = acc[v];
    }
  }
  __syncthreads();

  // ---------------- Phase 2: T OAMP iterations ----------------------------
  for (int t = 0; t < T; ++t) {
    const float g  = ggam[t];
    const float th = gth[t];

    // A = v2 * HHt + sigma2 * I
    for (int i = tid; i < 4096; i += 256) {
      float a = v2 * sHHt[i];
      if ((i >> 6) == (i & 63)) a += sigma2;
      sA[i] = a;
    }
    __syncthreads();

    // In-place Cholesky (lower), right-looking
    {
      const int tx = tid & 63, ty = tid >> 6;
      for (int k = 0; k < 64; ++k) {
        if (tid == 0) {
          float d = sqrtf(sA[k * 64 + k]);
          sA[k * 64 + k] = d;
          sDi[k] = 1.f / d;
        }
        __syncthreads();
        const float dinv = sDi[k];
        for (int i = k + 1 + tid; i < 64; i += 256) sA[i * 64 + k] *= dinv;
        __syncthreads();
        for (int i = k + 1 + ty; i < 64; i += 4) {
          const float lik = sA[i * 64 + k];
          for (int j = k + 1 + tx; j <= i; j += 64)
            sA[i * 64 + j] -= lik * sA[j * 64 + k];
        }
        __syncthreads();
      }
    }

    // S = A^{-1} H  (two triangular solves; one column per lane of wave 0)
    for (int i = tid; i < 2048; i += 256) sS[i] = sH[i];
    __syncthreads();
    if (wave == 0) {
      const int c = lane;
      for (int i = 0; i < 64; ++i) {            // L z = H
        float acc = sS[i * 32 + c];
        for (int k = 0; k < i; ++k) acc -= sA[i * 64 + k] * sS[k * 32 + c];
        sS[i * 32 + c] = acc * sDi[i];
      }
      for (int i = 63; i >= 0; --i) {           // L^T s = z
        float acc = sS[i * 32 + c];
        for (int k = i + 1; k < 64; ++k) acc -= sA[k * 64 + i] * sS[k * 32 + c];
        sS[i * 32 + c] = acc * sDi[i];
      }
    }
    __syncthreads();

    // ip = sum(S o H)  (v2 cancels: W = (32/ip) S^T),  ss = ||S||_F^2
    if (tid == 0) { sred[0] = 0.f; sred[1] = 0.f; sred[2] = 0.f; sred[3] = 0.f; }
    __syncthreads();
    float ip = 0.f, ss = 0.f;
    for (int i = tid; i < 2048; i += 256) {
      const float s = sS[i];
      ip += s * sH[i];
      ss += s * s;
    }
    ip = waveReduce(ip);
    ss = waveReduce(ss);
    if (lane == 0) { atomicAdd(&sred[0], ip); atomicAdd(&sred[1], ss); }
    __syncthreads();
    const float alpha = 32.f / sred[0];
    const float ssum  = sred[1];

    // z = y - H x
    if (tid < 64) {
      float acc = sy[tid];
      for (int n = 0; n < 32; ++n) acc -= sH[tid * 32 + n] * sx[n];
      sz[tid] = acc;
    }
    __syncthreads();

    // r = x + g * alpha * (S^T z)      (wave 0)
    if (wave == 0) {
      float acc = 0.f;
      for (int m = 0; m < 64; ++m) acc += sS[m * 32 + lane] * sz[m];
      sr[lane] = sx[lane] + g * alpha * acc;
    }

    // G = S^T H via fp32 WMMA (2x2 tiles, waves 0..3); C = I - g*alpha*G
    if (wave < 4) {
      const int ti = (wave >> 1) * 16;
      const int tj = (wave & 1) * 16;
      const int m  = lane & 15;
      const int kh = (lane >> 4) * 2;
      v8f acc = {};
      for (int k0 = 0; k0 < 64; k0 += 4) {
        v2f a, bb;
        a.x  = sS[(k0 + kh + 0) * 32 + ti + m];   // A[m][k] = S[k][ti+m]
        a.y  = sS[(k0 + kh + 1) * 32 + ti + m];
        bb.x = sH[(k0 + kh + 0) * 32 + tj + m];   // B[k][n] = H[k][tj+n]
        bb.y = sH[(k0 + kh + 1) * 32 + tj + m];
        acc = wmma_f32_4(a, bb, acc);
      }
      const int half = lane >> 4, n = lane & 15;
      #pragma unroll
      for (int v = 0; v < 8; ++v)
        sG[(ti + v + half * 8) * 32 + tj + n] = acc[v];
    }
    __syncthreads();

    // tau2 = (v2 * ||C||_F^2 + th^2 * sigma2 * ||W||_F^2) / 32
    const float ga = g * alpha;
    float tcc = 0.f;
    for (int i = tid; i < 1024; i += 256) {
      const int n = i >> 5, j = i & 31;
      const float cij = ((n == j) ? 1.f : 0.f) - ga * sG[i];
      tcc += cij * cij;
    }
    tcc = waveReduce(tcc);
    if (lane == 0) atomicAdd(&sred[3], tcc);
    __syncthreads();
    const float trCCt = sred[3];
    const float trWWt = alpha * alpha * ssum;
    const float tau2 =
        fmaxf((v2 * trCCt + th * th * sigma2 * trWWt) / 32.f, EPSV);

    // BPSK MMSE denoiser
    if (tid < 32) sx[tid] = tanhf(sr[tid] / tau2);
    __syncthreads();

    // v2 = max((||y - H x||^2 - M sigma2) / trHtH, EPS)
    float rn = 0.f;
    if (tid < 64) {
      float acc = sy[tid];
      for (int n = 0; n < 32; ++n) acc -= sH[tid * 32 + n] * sx[n];
      rn = acc * acc;
    }
    rn = waveReduce(rn);
    if (lane == 0 && wave < 2) atomicAdd(&sred[2], rn);
    __syncthreads();
    v2 = fmaxf((sred[2] - 64.f * sigma2) / trHtH, EPSV);
  }

  if (tid < 32) gx[(size_t)b * 32 + tid] = sx[tid];
}

extern "C" void kernel_launch(void* const* d_in, const int* in_sizes, int n_in,
                              void* d_out, int out_size, void* d_ws,
                              size_t ws_size, hipStream_t stream) {
  (void)n_in; (void)out_size; (void)d_ws; (void)ws_size;
  const float* H   = (const float*)d_in[0];
  const float* y   = (const float*)d_in[1];
  const float* sig = (const float*)d_in[2];
  const float* gam = (const float*)d_in[3];
  const float* th  = (const float*)d_in[4];
  float* out = (float*)d_out;
  const int B = in_sizes[2];   // 8192 batches (sigma2 has one entry per batch)
  const int T = in_sizes[3];   // 10 iterations (gamma length)
  oampnet_kernel<<<dim3(B), dim3(256), 0, stream>>>(H, y, sig, gam, th, out, T);
}